// KGETCDA_61203283968808
// MI455X (gfx1250) — compile-verified
//
#include <hip/hip_runtime.h>
#include <hip/hip_bf16.h>

#define D 96
#define XSTR 97            // LDS row stride (odd -> conflict-free column reads)
#define NODES_PER_BLOCK 64 // 4 waves x 16-node tiles
#define NEG_SLOPE 0.01f

typedef float v2f __attribute__((ext_vector_type(2)));
typedef float v8f __attribute__((ext_vector_type(8)));

// ---------------- zero the scatter target (side) ----------------
__global__ void kgetcda_zero(float* __restrict__ side, int total) {
    int i = blockIdx.x * blockDim.x + threadIdx.x;
    if (i < total) side[i] = 0.0f;
}

// ---------------- COO SpMM: side[row] += val * ego[col] ----------------
// one wave32 per edge; lane l handles features l, l+32, l+64
__global__ void kgetcda_spmm(const float* __restrict__ ego,
                             const float* __restrict__ vals,
                             const int*   __restrict__ rows,
                             const int*   __restrict__ cols,
                             float*       __restrict__ side,
                             int E) {
    int tid  = blockIdx.x * blockDim.x + threadIdx.x;
    int e    = tid >> 5;
    int lane = tid & 31;
    if (e >= E) return;
    int   r = rows[e];
    int   c = cols[e];
    float v = vals[e];
    const float* src = ego  + (size_t)c * D;
    float*       dst = side + (size_t)r * D;
#pragma unroll
    for (int k = lane; k < D; k += 32) {
        atomicAdd(&dst[k], v * src[k]);   // global_atomic_add_f32, non-returning
    }
}

// ---------------- fused bi-interaction: WMMA f32 16x16x4 ----------------
// out = leaky((ego+side) @ W1^T + b1) + leaky((ego*side) @ W2^T + b2)
__global__ void __launch_bounds__(128)
kgetcda_fused(const float* __restrict__ ego,  const float* __restrict__ side,
              const float* __restrict__ W1,   const float* __restrict__ b1,
              const float* __restrict__ W2,   const float* __restrict__ b2,
              float* __restrict__ out, int N) {
    __shared__ float sXs[NODES_PER_BLOCK * XSTR]; // ego + side
    __shared__ float sXm[NODES_PER_BLOCK * XSTR]; // ego * side

    const int tid      = threadIdx.x;
    const int nodeBase = blockIdx.x * NODES_PER_BLOCK;

    // Stage both interaction matrices into LDS (coalesced global reads).
    for (int i = tid; i < NODES_PER_BLOCK * D; i += 128) {
        int ln = i / D;
        int k  = i - ln * D;
        int node = nodeBase + ln;
        if (node < N) {
            float e = ego [(size_t)node * D + k];
            float s = side[(size_t)node * D + k];
            sXs[ln * XSTR + k] = e + s;
            sXm[ln * XSTR + k] = e * s;
        }
    }
    __syncthreads();

    const int wave = tid >> 5;
    const int lane = tid & 31;
    const int tileNode = nodeBase + wave * 16;

    // N is a multiple of 16 -> this branch is wave-uniform; EXEC all-ones
    // inside the WMMA region as required by the ISA.
    if (tileNode + 15 < N || tileNode < N) {
        if (tileNode >= N) return;
        const int n = lane & 15;   // output column within 16-tile / B col
        const int g = lane >> 4;   // half-wave selector (K pair)
        const int m = lane & 15;   // A-fragment row

        const float* xs = &sXs[(wave * 16) * XSTR];
        const float* xm = &sXm[(wave * 16) * XSTR];

        for (int j = 0; j < D; j += 16) {
            v8f acc1, acc2;
            const float bb1 = b1[j + n];
            const float bb2 = b2[j + n];
#pragma unroll
            for (int r = 0; r < 8; ++r) { acc1[r] = bb1; acc2[r] = bb2; }

            const float* w1row = W1 + (size_t)(j + n) * D; // B[k][n] = W[j+n][k]
            const float* w2row = W2 + (size_t)(j + n) * D;

#pragma unroll
            for (int kb = 0; kb < D; kb += 4) {
                const int kk = kb + 2 * g; // lanes 0-15: K0/K1, lanes 16-31: K2/K3
                v2f a1, a2, bf1, bf2;
                a1.x  = xs[m * XSTR + kk];     a1.y  = xs[m * XSTR + kk + 1];
                a2.x  = xm[m * XSTR + kk];     a2.y  = xm[m * XSTR + kk + 1];
                bf1.x = w1row[kk];             bf1.y = w1row[kk + 1];
                bf2.x = w2row[kk];             bf2.y = w2row[kk + 1];
                acc1 = __builtin_amdgcn_wmma_f32_16x16x4_f32(
                           false, a1, false, bf1, (short)0, acc1, false, false);
                acc2 = __builtin_amdgcn_wmma_f32_16x16x4_f32(
                           false, a2, false, bf2, (short)0, acc2, false, false);
            }

            // LeakyReLU both, sum, store (C/D layout: row = r + 8*g, col = n)
#pragma unroll
            for (int r = 0; r < 8; ++r) {
                float x1 = acc1[r]; x1 = (x1 >= 0.0f) ? x1 : NEG_SLOPE * x1;
                float x2 = acc2[r]; x2 = (x2 >= 0.0f) ? x2 : NEG_SLOPE * x2;
                int row = r + 8 * g;
                out[(size_t)(tileNode + row) * D + j + n] = x1 + x2;
            }
        }
    }
}

extern "C" void kernel_launch(void* const* d_in, const int* in_sizes, int n_in,
                              void* d_out, int out_size, void* d_ws, size_t ws_size,
                              hipStream_t stream) {
    const float* ego    = (const float*)d_in[0];
    const float* a_vals = (const float*)d_in[1];
    const float* W1     = (const float*)d_in[2];
    const float* b1     = (const float*)d_in[3];
    const float* W2     = (const float*)d_in[4];
    const float* b2     = (const float*)d_in[5];
    const int*   a_rows = (const int*)d_in[6];
    const int*   a_cols = (const int*)d_in[7];
    float* out  = (float*)d_out;
    float* side = (float*)d_ws;          // N*D floats of scratch

    const int N = in_sizes[0] / D;       // 50000
    const int E = in_sizes[6];           // 800000

    // 1) zero the scatter buffer
    const int total = N * D;
    kgetcda_zero<<<(total + 255) / 256, 256, 0, stream>>>(side, total);

    // 2) COO SpMM scatter: one wave per edge (8 edges / 256-thread block)
    const int spmmBlocks = (E + 7) / 8;
    kgetcda_spmm<<<spmmBlocks, 256, 0, stream>>>(ego, a_vals, a_rows, a_cols,
                                                 side, E);

    // 3) fused WMMA bi-interaction aggregator
    const int fusedBlocks = (N + NODES_PER_BLOCK - 1) / NODES_PER_BLOCK;
    kgetcda_fused<<<fusedBlocks, 128, 0, stream>>>(ego, side, W1, b1, W2, b2,
                                                   out, N);
}